// SimplicialConvolution_37718402793517
// MI455X (gfx1250) — compile-verified
//
#include <hip/hip_runtime.h>

#define BN   8
#define CIN  32
#define COUT 64
#define KC   4
#define F    256   // BN*CIN, feature width of the batched node matrix

typedef __attribute__((ext_vector_type(2))) float v2f;
typedef __attribute__((ext_vector_type(8))) float v8f;

// ---------- x (B,CIN,M) -> T0 (M, F) via LDS tiled transpose ----------
__global__ void k_transpose_x(const float* __restrict__ x, float* __restrict__ t0, int M) {
    __shared__ float tile[32][33];
    int mBase = blockIdx.x * 32;
    int cBase = blockIdx.y * 32;   // channel = b*CIN + i, 0..255
    int tx = threadIdx.x, ty = threadIdx.y;
#pragma unroll
    for (int i = 0; i < 32; i += 8) {
        int c = cBase + ty + i;
        int m = mBase + tx;
        if (m < M) tile[ty + i][tx] = x[(size_t)c * M + m];
    }
    __syncthreads();
#pragma unroll
    for (int i = 0; i < 32; i += 8) {
        int m = mBase + ty + i;
        int c = cBase + tx;
        if (m < M) t0[(size_t)m * F + c] = tile[tx][ty + i];
    }
}

// ---------- dst = alpha*src (src==nullptr -> 0) ----------
__global__ void k_scale_init(float* __restrict__ dst, const float* __restrict__ src,
                             float alpha, size_t n) {
    size_t i = (size_t)blockIdx.x * blockDim.x + threadIdx.x;
    size_t stride = (size_t)gridDim.x * blockDim.x;
    for (; i < n; i += stride) dst[i] = src ? alpha * src[i] : 0.0f;
}

// ---------- dst[row,:] += scale*val * src[col,:]  (one wave per nnz) ----------
__global__ void k_spmm_atomic(const int* __restrict__ rows, const int* __restrict__ cols,
                              const float* __restrict__ vals, const float* __restrict__ src,
                              float* __restrict__ dst, float scale, int nnz) {
    int e = blockIdx.x * (blockDim.x >> 5) + (threadIdx.x >> 5);
    if (e >= nnz) return;
    int lane = threadIdx.x & 31;
    int r = rows[e];
    int c = cols[e];
    float v = vals[e] * scale;
    const float* s = src + (size_t)c * F + lane * 4;
    float*       d = dst + (size_t)r * F + lane * 4;
    float4 a0 = *(const float4*)(s);
    float4 a1 = *(const float4*)(s + 128);
    atomicAdd(d + 0,       v * a0.x);
    atomicAdd(d + 1,       v * a0.y);
    atomicAdd(d + 2,       v * a0.z);
    atomicAdd(d + 3,       v * a0.w);
    atomicAdd(d + 128 + 0, v * a1.x);
    atomicAdd(d + 128 + 1, v * a1.y);
    atomicAdd(d + 128 + 2, v * a1.z);
    atomicAdd(d + 128 + 3, v * a1.w);
}

// ---------- dense contraction on the matrix pipe ----------
// A(16x4, f32) = theta tile rows=o, reduction q = k*32+i  (LDS, pair-interleaved)
// B(4x16,  f32) = X_k fragment, cols = m
// D(16x16, f32) accumulators, 4 row tiles over o -> 64 outputs, fixed b per block.y
__global__ void k_cheb_wmma(const float* __restrict__ X,      // (KC, M, F) slices
                            const float* __restrict__ theta,  // (COUT, CIN, KC)
                            const float* __restrict__ bias,   // (COUT)
                            float* __restrict__ out,          // (BN, COUT, M)
                            int M) {
    // Pair-interleaved layout so each A-fragment is one contiguous 8-byte LDS load:
    // sA[(q>>1)*128 + o*2 + (q&1)] = theta[o][i][k],  q = k*32 + i   (32 KB)
    __shared__ float sA[KC * CIN * COUT];
    for (int t = threadIdx.x; t < KC * CIN * COUT; t += blockDim.x) {
        int d  = t & 1;
        int o  = (t >> 1) & 63;
        int qq = t >> 7;
        int q  = 2 * qq + d;
        int i  = q & 31;
        int k  = q >> 5;
        sA[t] = theta[(size_t)o * (CIN * KC) + i * KC + k];
    }
    __syncthreads();

    int wave = threadIdx.x >> 5;
    int lane = threadIdx.x & 31;
    int tilem = blockIdx.x * (blockDim.x >> 5) + wave;
    int nTiles = M >> 4;                    // M divisible by 16 (100000/16 = 6250)
    if (tilem >= nTiles) return;            // whole-wave uniform -> EXEC stays all-ones
    int m0   = tilem << 4;
    int b    = blockIdx.y;
    int half = lane >> 4;                   // 0: K=q+0,q+1   1: K=q+2,q+3  (ISA 16x4 layout)
    int l    = lane & 15;

    v8f acc[4] = {v8f{}, v8f{}, v8f{}, v8f{}};

    // B-fragment base: X_k[m0+l][b*32 + i0 + half*2 .. +1] -> one float2 load per step
    const float* xr = X + (size_t)(m0 + l) * F + b * CIN + half * 2;

#pragma unroll 2
    for (int kk = 0; kk < 32; ++kk) {
        int q      = kk * 4;
        int kslice = q >> 5;
        int i0     = q & 31;
        v2f bf = *(const v2f*)(xr + (size_t)kslice * M * F + i0);
        if (kk + 1 < 32) {
            int qn = (kk + 1) * 4;
            __builtin_prefetch(xr + (size_t)(qn >> 5) * M * F + (qn & 31), 0, 3);
        }
        // A-frag: rows q+half*2, q+half*2+1 are adjacent in the pair-interleaved layout
        const float* ap = &sA[((q + half * 2) >> 1) * (2 * COUT) + l * 2];
#pragma unroll
        for (int ot = 0; ot < 4; ++ot) {
            v2f af = *(const v2f*)(ap + ot * 32);   // o = ot*16 + l
            acc[ot] = __builtin_amdgcn_wmma_f32_16x16x4_f32(
                false, af, false, bf, (short)0, acc[ot], false, false);
        }
    }

    // D layout: lanes 0-15 N=l M=j ; lanes 16-31 N=l M=j+8.  N->m, M->o.
#pragma unroll
    for (int ot = 0; ot < 4; ++ot) {
#pragma unroll
        for (int j = 0; j < 8; ++j) {
            int o = ot * 16 + j + half * 8;
            out[((size_t)b * COUT + o) * M + m0 + l] = acc[ot][j] + bias[o];
        }
    }
}

extern "C" void kernel_launch(void* const* d_in, const int* in_sizes, int n_in,
                              void* d_out, int out_size, void* d_ws, size_t ws_size,
                              hipStream_t stream) {
    const float* x     = (const float*)d_in[0];
    const int*   L_row = (const int*)d_in[1];
    const int*   L_col = (const int*)d_in[2];
    const float* L_val = (const float*)d_in[3];
    const float* theta = (const float*)d_in[4];
    const float* bias  = (const float*)d_in[5];
    float* out = (float*)d_out;

    int M   = in_sizes[0] / F;    // 100000
    int nnz = in_sizes[1];        // 1.6M
    size_t sliceElems = (size_t)M * F;   // 25.6M floats per Chebyshev slice

    // ws layout: 4 Chebyshev slices T0..T3 = 409.6 MB
    float* T0 = (float*)d_ws;
    float* T1 = T0 + sliceElems;
    float* T2 = T1 + sliceElems;
    float* T3 = T2 + sliceElems;

    // 1) T0 = transpose(x)
    dim3 tb(32, 8);
    dim3 tg((M + 31) / 32, F / 32);
    k_transpose_x<<<tg, tb, 0, stream>>>(x, T0, M);

    int wavesPerBlock = 8;
    int spmmBlocks = (nnz + wavesPerBlock - 1) / wavesPerBlock;

    // 2) T1 = L @ T0
    k_scale_init<<<4096, 256, 0, stream>>>(T1, nullptr, 0.0f, sliceElems);
    k_spmm_atomic<<<spmmBlocks, 256, 0, stream>>>(L_row, L_col, L_val, T0, T1, 1.0f, nnz);

    // 3) T2 = 2*L@T1 - T0
    k_scale_init<<<4096, 256, 0, stream>>>(T2, T0, -1.0f, sliceElems);
    k_spmm_atomic<<<spmmBlocks, 256, 0, stream>>>(L_row, L_col, L_val, T1, T2, 2.0f, nnz);

    // 4) T3 = 2*L@T2 - T1
    k_scale_init<<<4096, 256, 0, stream>>>(T3, T1, -1.0f, sliceElems);
    k_spmm_atomic<<<spmmBlocks, 256, 0, stream>>>(L_row, L_col, L_val, T2, T3, 2.0f, nnz);

    // 5) y = einsum + bias on the WMMA pipe (full f32)
    int nTiles = M / 16;
    dim3 gg((nTiles + wavesPerBlock - 1) / wavesPerBlock, BN);
    k_cheb_wmma<<<gg, 256, 0, stream>>>(T0, theta, bias, out, M);
}